// HybridGATGCN_29222957481999
// MI455X (gfx1250) — compile-verified
//
#include <hip/hip_runtime.h>
#include <hip/hip_bf16.h>
#include <math.h>

typedef __attribute__((ext_vector_type(2))) float v2f;
typedef __attribute__((ext_vector_type(8))) float v8f;

#define F_IN  64
#define H1    64
#define H2    64
#define HEADS 4
#define HC    256   // HEADS*H2
#define OUTF  32

#define CDIV(a,b) (((a)+(b)-1)/(b))

// ---------------- utility ----------------
__global__ void fill_kernel(float* __restrict__ p, float v, long n) {
  long i = (long)blockIdx.x * blockDim.x + threadIdx.x;
  if (i < n) p[i] = v;
}

// ---------------- fp32 WMMA GEMM: C[M,Nc] = A[M,K] @ B[K,Nc] ----------------
// one wave per 16x16 output tile; V_WMMA_F32_16X16X4_F32, K stepped by 4.
__global__ void wmma_gemm_f32(const float* __restrict__ A, const float* __restrict__ B,
                              float* __restrict__ C, int M, int K, int Nc) {
  int gtid   = blockIdx.x * blockDim.x + threadIdx.x;
  int wave   = gtid >> 5;
  int lane   = threadIdx.x & 31;
  int ntiles = Nc >> 4;
  int mt = wave / ntiles;
  int nt = wave - mt * ntiles;
  if (mt * 16 >= M) return;                 // wave-uniform exit, EXEC all-ones inside
  int m0 = mt << 4, n0 = nt << 4;
  int half = lane >> 4;                     // 0: K pair {0,1}, 1: K pair {2,3}
  int l    = lane & 15;

  v8f acc = {};
  const float* arow = A + (size_t)(m0 + l) * K + (half << 1);
  const float* bcol = B + (size_t)(half << 1) * Nc + n0 + l;
  for (int kk = 0; kk < K; kk += 4) {
    v2f a; a.x = arow[kk];              a.y = arow[kk + 1];
    v2f b; b.x = bcol[(size_t)kk * Nc]; b.y = bcol[(size_t)(kk + 1) * Nc];
    acc = __builtin_amdgcn_wmma_f32_16x16x4_f32(false, a, false, b,
                                                (short)0, acc, false, false);
  }
  // C/D layout: VGPR v -> M = m0 + v + half*8, N = n0 + l
  float* cp = C + (size_t)(m0 + half * 8) * Nc + n0 + l;
  #pragma unroll
  for (int v = 0; v < 8; ++v) cp[(size_t)v * Nc] = acc[v];
}

// ---------------- GCN ----------------
__global__ void deg_kernel(const int* __restrict__ ei, int E, float* __restrict__ deg) {
  int e = blockIdx.x * blockDim.x + threadIdx.x;
  if (e < E) atomicAdd(&deg[ei[E + e]], 1.0f);
}

__global__ void rsqrt_kernel(float* __restrict__ deg, int n) {
  int i = blockIdx.x * blockDim.x + threadIdx.x;
  if (i < n) deg[i] = rsqrtf(deg[i]);       // deg >= 1 (self-loop) always
}

// one wave per edge (incl. self-loops), each lane scatters 2 of 64 channels
__global__ void gcn_scatter(const int* __restrict__ ei, int E, int N,
                            const float* __restrict__ dinv,
                            const float* __restrict__ hin, float* __restrict__ hout) {
  long gtid = (long)blockIdx.x * blockDim.x + threadIdx.x;
  long e    = gtid >> 5;
  int lane  = threadIdx.x & 31;
  long EP   = (long)E + N;
  if (e >= EP) return;
  int s, d;
  if (e < E) { s = ei[e]; d = ei[E + e]; } else { s = d = (int)(e - E); }
  float norm = dinv[s] * dinv[d];
  float2 v = ((const float2*)(hin + (size_t)s * H1))[lane];
  float* dp = hout + (size_t)d * H1 + lane * 2;
  atomicAdd(dp + 0, norm * v.x);
  atomicAdd(dp + 1, norm * v.y);
}

__global__ void bias_relu_kernel(float* __restrict__ h, const float* __restrict__ b,
                                 long n, int cmask) {
  long i = (long)blockIdx.x * blockDim.x + threadIdx.x;
  if (i < n) { float v = h[i] + b[i & cmask]; h[i] = v > 0.0f ? v : 0.0f; }
}

// ---------------- GAT attention ----------------
__global__ void att_kernel(const float* __restrict__ g, const float* __restrict__ att_src,
                           const float* __restrict__ att_dst,
                           float* __restrict__ a_s, float* __restrict__ a_d, int N) {
  int idx = blockIdx.x * blockDim.x + threadIdx.x;   // node*HEADS + head
  if (idx >= N * HEADS) return;
  int hh = idx & 3, n = idx >> 2;
  const float* gp = g + (size_t)n * HC + hh * H2;
  const float* ws = att_src + hh * H2;
  const float* wd = att_dst + hh * H2;
  float ss = 0.0f, sd = 0.0f;
  #pragma unroll 8
  for (int c = 0; c < H2; ++c) { float gv = gp[c]; ss += gv * ws[c]; sd += gv * wd[c]; }
  a_s[idx] = ss; a_d[idx] = sd;
}

__global__ void edge_max_kernel(const int* __restrict__ ei, int E, int N,
                                const float* __restrict__ a_s, const float* __restrict__ a_d,
                                float* __restrict__ ebuf, float* __restrict__ m) {
  long idx = (long)blockIdx.x * blockDim.x + threadIdx.x;   // edge*HEADS + head
  long EP  = (long)E + N;
  if (idx >= EP * HEADS) return;
  long e = idx >> 2; int hh = (int)(idx & 3);
  int s, d;
  if (e < E) { s = ei[e]; d = ei[E + e]; } else { s = d = (int)(e - E); }
  float v = a_s[(size_t)s * HEADS + hh] + a_d[(size_t)d * HEADS + hh];
  v = v > 0.0f ? v : 0.2f * v;                              // leaky_relu(0.2)
  ebuf[idx] = v;
  atomicMax(&m[(size_t)d * HEADS + hh], v);                 // global_atomic_max_num_f32
}

__global__ void edge_exp_kernel(const int* __restrict__ ei, int E, int N,
                                const float* __restrict__ m,
                                float* __restrict__ ebuf, float* __restrict__ denom) {
  long idx = (long)blockIdx.x * blockDim.x + threadIdx.x;
  long EP  = (long)E + N;
  if (idx >= EP * HEADS) return;
  long e = idx >> 2; int hh = (int)(idx & 3);
  int d = (e < E) ? ei[E + e] : (int)(e - E);
  float w = __expf(ebuf[idx] - m[(size_t)d * HEADS + hh]);
  ebuf[idx] = w;
  atomicAdd(&denom[(size_t)d * HEADS + hh], w);
}

// one wave per edge; lane covers 8 consecutive channels of 256 (head = lane/8)
__global__ void gat_scatter(const int* __restrict__ ei, int E, int N,
                            const float* __restrict__ ebuf, const float* __restrict__ denom,
                            const float* __restrict__ g, float* __restrict__ agg) {
  long gtid = (long)blockIdx.x * blockDim.x + threadIdx.x;
  long e    = gtid >> 5;
  int lane  = threadIdx.x & 31;
  long EP   = (long)E + N;
  if (e >= EP) return;
  int s, d;
  if (e < E) { s = ei[e]; d = ei[E + e]; } else { s = d = (int)(e - E); }
  int base = lane * 8;
  int hh   = lane >> 3;
  float alpha = ebuf[e * HEADS + hh] / denom[(size_t)d * HEADS + hh];
  const float4* sp = (const float4*)(g + (size_t)s * HC + base);
  float4 v0 = sp[0], v1 = sp[1];
  float* dp = agg + (size_t)d * HC + base;
  atomicAdd(dp + 0, alpha * v0.x); atomicAdd(dp + 1, alpha * v0.y);
  atomicAdd(dp + 2, alpha * v0.z); atomicAdd(dp + 3, alpha * v0.w);
  atomicAdd(dp + 4, alpha * v1.x); atomicAdd(dp + 5, alpha * v1.y);
  atomicAdd(dp + 6, alpha * v1.z); atomicAdd(dp + 7, alpha * v1.w);
}

// ---------------- pooling + output ----------------
__global__ void pool_kernel(const float* __restrict__ agg, const float* __restrict__ gat_b,
                            const int* __restrict__ batch, float* __restrict__ pooled, int N) {
  long i = (long)blockIdx.x * blockDim.x + threadIdx.x;
  if (i >= (long)N * HC) return;
  int  c = (int)(i & (HC - 1));
  long n = i >> 8;
  float z = agg[i] + gat_b[c];
  z = z > 0.0f ? z : 0.0f;
  atomicAdd(&pooled[(size_t)batch[n] * HC + c], z);
}

__global__ void cnt_kernel(const int* __restrict__ batch, float* __restrict__ cnt, int N) {
  int n = blockIdx.x * blockDim.x + threadIdx.x;
  if (n < N) atomicAdd(&cnt[batch[n]], 1.0f);
}

__global__ void out_kernel(const float* __restrict__ pooled, const float* __restrict__ cnt,
                           const float* __restrict__ W, const float* __restrict__ b,
                           float* __restrict__ out, int G) {
  int idx = blockIdx.x * blockDim.x + threadIdx.x;   // graph*OUTF + o
  if (idx >= G * OUTF) return;
  int o  = idx & (OUTF - 1);
  int gi = idx >> 5;
  float inv = 1.0f / fmaxf(cnt[gi], 1.0f);
  const float* pp = pooled + (size_t)gi * HC;
  float acc = b[o];
  for (int c = 0; c < HC; ++c) acc += pp[c] * inv * W[c * OUTF + o];
  out[idx] = acc;
}

// ---------------- launch ----------------
extern "C" void kernel_launch(void* const* d_in, const int* in_sizes, int n_in,
                              void* d_out, int out_size, void* d_ws, size_t ws_size,
                              hipStream_t stream) {
  const float* x        = (const float*)d_in[0];
  const int*   ei       = (const int*)  d_in[1];
  const int*   batch    = (const int*)  d_in[2];
  const float* gcn_W    = (const float*)d_in[3];
  const float* gcn_b    = (const float*)d_in[4];
  const float* gat_W    = (const float*)d_in[5];
  const float* att_src  = (const float*)d_in[6];
  const float* att_dst  = (const float*)d_in[7];
  const float* gat_b    = (const float*)d_in[8];
  const float* out_W    = (const float*)d_in[9];
  const float* out_b    = (const float*)d_in[10];

  const int N = in_sizes[0] / F_IN;
  const int E = in_sizes[1] / 2;
  const int G = out_size / OUTF;
  const long EP = (long)E + N;

  // workspace layout (floats)
  float* ws = (float*)d_ws;
  float* h_tmp  = ws;                    ws += (size_t)N * H1;     // x @ gcn_W
  float* h      = ws;                    ws += (size_t)N * H1;     // after scatter+relu
  float* dinv   = ws;                    ws += (size_t)N;          // deg -> deg^-1/2
  float* g      = ws;                    ws += (size_t)N * HC;     // h @ gat_W
  float* a_s    = ws;                    ws += (size_t)N * HEADS;
  float* a_d    = ws;                    ws += (size_t)N * HEADS;
  float* ebuf   = ws;                    ws += (size_t)EP * HEADS; // edge logits/weights
  float* segm   = ws;                    ws += (size_t)N * HEADS;  // segment max
  float* denom  = ws;                    ws += (size_t)N * HEADS;
  float* agg    = ws;                    ws += (size_t)N * HC;
  float* pooled = ws;                    ws += (size_t)G * HC;
  float* cnt    = ws;                    ws += (size_t)G;

  const int T = 256;

  // ---- GCN ----
  fill_kernel<<<CDIV((long)N, T), T, 0, stream>>>(dinv, 1.0f, (long)N);        // self-loop deg
  fill_kernel<<<CDIV((long)N * H1, T), T, 0, stream>>>(h, 0.0f, (long)N * H1);

  { // h_tmp = x @ gcn_W  (WMMA fp32)
    long waves = (long)(N / 16) * (H1 / 16);
    wmma_gemm_f32<<<CDIV(waves * 32, (long)T), T, 0, stream>>>(x, gcn_W, h_tmp, N, F_IN, H1);
  }
  deg_kernel<<<CDIV(E, T), T, 0, stream>>>(ei, E, dinv);
  rsqrt_kernel<<<CDIV(N, T), T, 0, stream>>>(dinv, N);
  gcn_scatter<<<CDIV(EP * 32, (long)T), T, 0, stream>>>(ei, E, N, dinv, h_tmp, h);
  bias_relu_kernel<<<CDIV((long)N * H1, T), T, 0, stream>>>(h, gcn_b, (long)N * H1, H1 - 1);

  // ---- GAT ----
  { // g = h @ gat_W  (WMMA fp32)
    long waves = (long)(N / 16) * (HC / 16);
    wmma_gemm_f32<<<CDIV(waves * 32, (long)T), T, 0, stream>>>(h, gat_W, g, N, H1, HC);
  }
  att_kernel<<<CDIV(N * HEADS, T), T, 0, stream>>>(g, att_src, att_dst, a_s, a_d, N);

  fill_kernel<<<CDIV((long)N * HEADS, T), T, 0, stream>>>(segm, -1e30f, (long)N * HEADS);
  fill_kernel<<<CDIV((long)N * HEADS, T), T, 0, stream>>>(denom, 0.0f, (long)N * HEADS);
  fill_kernel<<<CDIV((long)N * HC, T), T, 0, stream>>>(agg, 0.0f, (long)N * HC);

  edge_max_kernel<<<CDIV(EP * HEADS, (long)T), T, 0, stream>>>(ei, E, N, a_s, a_d, ebuf, segm);
  edge_exp_kernel<<<CDIV(EP * HEADS, (long)T), T, 0, stream>>>(ei, E, N, segm, ebuf, denom);
  gat_scatter<<<CDIV(EP * 32, (long)T), T, 0, stream>>>(ei, E, N, ebuf, denom, g, agg);

  // ---- pool + output ----
  fill_kernel<<<CDIV((long)G * HC, T), T, 0, stream>>>(pooled, 0.0f, (long)G * HC);
  fill_kernel<<<CDIV((long)G, T), T, 0, stream>>>(cnt, 0.0f, (long)G);
  pool_kernel<<<CDIV((long)N * HC, T), T, 0, stream>>>(agg, gat_b, batch, pooled, N);
  cnt_kernel<<<CDIV(N, T), T, 0, stream>>>(batch, cnt, N);
  out_kernel<<<CDIV(G * OUTF, T), T, 0, stream>>>(pooled, cnt, out_W, out_b, (float*)d_out, G);
}